// Pooling_37271726195210
// MI455X (gfx1250) — compile-verified
//
#include <hip/hip_runtime.h>
#include <hip/hip_bf16.h>
#include <math.h>

#define B_DIM 8
#define S_DIM 8192
#define H_DIM 1024
#define K_TOP 32

typedef __attribute__((ext_vector_type(2))) float v2f;
typedef __attribute__((ext_vector_type(8))) float v8f;

// ---------------------------------------------------------------------------
// Kernel 0: scale = 1 / (||w|| + 1e-6)
// ---------------------------------------------------------------------------
__global__ void wnorm_kernel(const float* __restrict__ w, float* __restrict__ scale_out) {
    __shared__ float red[256];
    float s = 0.f;
    for (int i = threadIdx.x; i < H_DIM; i += 256) {
        float v = w[i];
        s += v * v;
    }
    red[threadIdx.x] = s;
    __syncthreads();
    for (int off = 128; off > 0; off >>= 1) {
        if (threadIdx.x < off) red[threadIdx.x] += red[threadIdx.x + off];
        __syncthreads();
    }
    if (threadIdx.x == 0) {
        scale_out[0] = 1.f / (sqrtf(red[0]) + 1e-6f);
    }
}

// ---------------------------------------------------------------------------
// Kernel 1: scores via V_WMMA_F32_16X16X4_F32.
// Each wave32 owns 16 consecutive rows of x ([B*S, H]); loops K in steps of 4.
// A layout (32-bit A 16x4): VGPR0 = K0(lanes0-15)/K2(lanes16-31), VGPR1 = K1/K3.
// B = scaled w replicated across all 16 columns (K striping mirrors A:
// vgpr0 -> K{0,2}, vgpr1 -> K{1,3} by lane half), so every D column holds the
// dot products. D layout: vgpr v, lo half -> row v, hi half -> row v+8.
// ---------------------------------------------------------------------------
__global__ void score_kernel(const float* __restrict__ x,
                             const unsigned char* __restrict__ mask,
                             const float* __restrict__ wscale_ptr,  // pre-scaled? no: raw w below
                             const float* __restrict__ w,
                             float* __restrict__ y) {
    __shared__ float ws_lds[H_DIM];

    const float scale = wscale_ptr[0];
    for (int i = threadIdx.x; i < H_DIM; i += blockDim.x) {
        ws_lds[i] = w[i] * scale;
    }
    __syncthreads();

    const int wave = threadIdx.x >> 5;
    const int lane = threadIdx.x & 31;
    const int half = lane >> 4;       // 0: lanes 0-15, 1: lanes 16-31
    const int m    = lane & 15;       // row within tile (A), col (B/D)
    const int tile = blockIdx.x * (blockDim.x >> 5) + wave;
    const int row_base = tile * 16;

    const float* __restrict__ xrow = x + (size_t)(row_base + m) * H_DIM;
    const int ko = 2 * half;          // K offset owned by this lane half

    v8f acc = {};
    #pragma unroll 4
    for (int k = 0; k < H_DIM; k += 4) {
        v2f a, b;
        a.x = xrow[k + ko];
        a.y = xrow[k + ko + 1];
        b.x = ws_lds[k + ko];
        b.y = ws_lds[k + ko + 1];
        acc = __builtin_amdgcn_wmma_f32_16x16x4_f32(
            /*neg_a=*/false, a, /*neg_b=*/false, b,
            /*c_mod=*/(short)0, acc, /*reuse_a=*/false, /*reuse_b=*/false);
    }

    // Extract: lane with m==0 in each half holds rows (half*8 + v) in acc[v].
    if (m == 0) {
        #pragma unroll
        for (int v = 0; v < 8; ++v) {
            const int r = row_base + half * 8 + v;
            float val = acc[v];
            val += mask[r] ? 0.f : -1e6f;
            y[r] = val;
        }
    }
}

// ---------------------------------------------------------------------------
// Kernel 2: per-batch top-32 (descending, lowest index wins ties, like
// jax.lax.top_k). One block per batch; scores cached in LDS; 32 iterative
// arg-max reductions over a 64-bit (value_bits | inverted_index) key.
// ---------------------------------------------------------------------------
__global__ void topk_kernel(const float* __restrict__ y,
                            float* __restrict__ tvals,
                            int* __restrict__ tidx) {
    const int b = blockIdx.x;
    const int tid = threadIdx.x;
    __shared__ float sc[S_DIM];                    // 32 KB
    __shared__ unsigned long long red[256];

    const float* __restrict__ yb = y + (size_t)b * S_DIM;
    for (int i = tid; i < S_DIM; i += 256) sc[i] = yb[i];
    __syncthreads();

    for (int kk = 0; kk < K_TOP; ++kk) {
        unsigned long long best = 0ull;
        for (int i = tid; i < S_DIM; i += 256) {
            const float v = sc[i];
            unsigned u = __float_as_uint(v);
            u = (u >> 31) ? ~u : (u | 0x80000000u);   // order-preserving map
            const unsigned long long key =
                ((unsigned long long)u << 32) | (unsigned)(S_DIM - 1 - i);
            if (key > best) best = key;
        }
        red[tid] = best;
        __syncthreads();
        for (int off = 128; off > 0; off >>= 1) {
            if (tid < off) {
                const unsigned long long o = red[tid + off];
                if (o > red[tid]) red[tid] = o;
            }
            __syncthreads();
        }
        if (tid == 0) {
            const unsigned long long k0 = red[0];
            const int i = S_DIM - 1 - (int)(k0 & 0xFFFFFFFFu);
            tvals[b * K_TOP + kk] = sc[i];
            tidx[b * K_TOP + kk] = i;
            sc[i] = -INFINITY;                         // exclude from next round
        }
        __syncthreads();
    }
}

// ---------------------------------------------------------------------------
// Kernel 3: out[b,k,:] = x[b, idx[b,k], :] * tanh(val[b,k]).
// One block per (b,k); float4 streaming copy.
// ---------------------------------------------------------------------------
__global__ void gather_kernel(const float* __restrict__ x,
                              const float* __restrict__ tvals,
                              const int* __restrict__ tidx,
                              float* __restrict__ out) {
    const int b = blockIdx.x / K_TOP;
    const int k = blockIdx.x % K_TOP;
    const int idx = tidx[b * K_TOP + k];
    const float t = tanhf(tvals[b * K_TOP + k]);

    const float4* __restrict__ src =
        (const float4*)(x + ((size_t)b * S_DIM + idx) * H_DIM);
    float4* __restrict__ dst =
        (float4*)(out + ((size_t)b * K_TOP + k) * H_DIM);

    for (int i = threadIdx.x; i < H_DIM / 4; i += blockDim.x) {
        float4 v = src[i];
        v.x *= t; v.y *= t; v.z *= t; v.w *= t;
        dst[i] = v;
    }
}

// ---------------------------------------------------------------------------
extern "C" void kernel_launch(void* const* d_in, const int* in_sizes, int n_in,
                              void* d_out, int out_size, void* d_ws, size_t ws_size,
                              hipStream_t stream) {
    const float* x = (const float*)d_in[0];              // [B, S, H] f32
    const float* w = (const float*)d_in[1];              // [1, H]    f32
    const unsigned char* mask = (const unsigned char*)d_in[2];  // [B, S] bool
    float* out = (float*)d_out;                          // [B, K, H] f32

    // Workspace layout (all 256-byte aligned enough for f32/int):
    //   [0]                 : float scale
    //   [256 ..)            : float scores [B*S]
    //   then                : float tvals  [B*K]
    //   then                : int   tidx   [B*K]
    float* scale = (float*)d_ws;
    float* y     = (float*)((char*)d_ws + 256);
    float* tvals = y + (size_t)B_DIM * S_DIM;
    int*   tidx  = (int*)(tvals + B_DIM * K_TOP);

    // 0) 1/(||w||+1e-6)
    wnorm_kernel<<<1, 256, 0, stream>>>(w, scale);

    // 1) WMMA scoring: 65536 rows -> 4096 16-row wave tiles, 8 waves/block.
    const int tiles = (B_DIM * S_DIM) / 16;
    score_kernel<<<tiles / 8, 256, 0, stream>>>(x, mask, scale, w, y);

    // 2) top-32 per batch
    topk_kernel<<<B_DIM, 256, 0, stream>>>(y, tvals, tidx);

    // 3) gather + tanh rescale
    gather_kernel<<<B_DIM * K_TOP, 256, 0, stream>>>(x, tvals, tidx, out);
}